// NeuralCDE_21835613733134
// MI455X (gfx1250) — compile-verified
//
#include <hip/hip_runtime.h>
#include <hip/hip_bf16.h>
#include <math.h>

#define TT      128
#define BB      32
#define CC      64
#define HH      128
#define OUTN    8
#define NSTEPS  64
#define NBLK    32          // workgroups in main kernel (one per WGP)
#define NTHR    256         // 8 waves of 32
#define HL      4           // h-columns owned per WG (128/32)
#define NCOLS   256         // W2 columns per WG (HL*CC)
#define HSTR    132         // padded LDS row stride (bank-conflict free)

typedef float v2f __attribute__((ext_vector_type(2)));
typedef float v8f __attribute__((ext_vector_type(8)));

__device__ __forceinline__ v8f wmma4(v2f a, v2f b, v8f c) {
  // V_WMMA_F32_16X16X4_F32: fp32 A(16x4) x B(4x16) + C(16x16)
  return __builtin_amdgcn_wmma_f32_16x16x4_f32(false, a, false, b, (short)0, c,
                                               false, false);
}

// ---------------------------------------------------------------------------
// Kernel 0: zero the grid-barrier state (must run before the persistent kernel)
// ---------------------------------------------------------------------------
__global__ void cde_init(unsigned* syncv) {
  if (threadIdx.x == 0) { syncv[0] = 0u; syncv[1] = 0u; }
}

// ---------------------------------------------------------------------------
// Kernel 1: natural cubic spline (uniform knots) -> dX/dt table at the 129
// half-step times t_k = k*(127/128), k = 0..128.  dXg layout: [129][B][C].
// 2048 independent Thomas solves; pivot sequence nd[] is data-independent and
// shared per block.
// ---------------------------------------------------------------------------
__global__ void cde_spline(const float* __restrict__ x, float* __restrict__ dXg) {
  __shared__ float nd[TT];
  const int tid = threadIdx.x;
  const int sid = blockIdx.x * blockDim.x + tid;   // 0..2047  (= b*64 + c)
  if (tid == 0) {
    float pd = 2.0f;
    nd[0] = pd;
    for (int ii = 1; ii < TT; ++ii) {
      float d = (ii < TT - 1) ? 4.0f : 2.0f;
      pd = d - 1.0f / pd;
      nd[ii] = pd;
    }
  }
  __syncthreads();
  const int b = sid >> 6;
  const int c = sid & 63;
  const float* xp = x + (size_t)b * TT * CC + c;
#define XV(t) xp[(size_t)(t) * CC]
  float nb[TT];                       // forward RHS, then knot derivatives y
  float pb = 3.0f * (XV(1) - XV(0));
  nb[0] = pb;
  for (int ii = 1; ii < TT; ++ii) {
    float rhs = (ii < TT - 1) ? 3.0f * (XV(ii + 1) - XV(ii - 1))
                              : 3.0f * (XV(TT - 1) - XV(TT - 2));
    float wv = 1.0f / nd[ii - 1];
    pb = rhs - wv * pb;
    nb[ii] = pb;
  }
  nb[TT - 1] = nb[TT - 1] / nd[TT - 1];
  for (int ii = TT - 2; ii >= 0; --ii)
    nb[ii] = (nb[ii] - nb[ii + 1]) / nd[ii];
  // evaluate spline derivative at the 129 half-step times
  const float hdt = 127.0f / 128.0f;
  for (int k = 0; k <= 2 * NSTEPS; ++k) {
    float t = (float)k * hdt;
    int idx = (int)t;
    if (idx > TT - 2) idx = TT - 2;
    float frac = t - (float)idx;
    float y0 = nb[idx], y1 = nb[idx + 1];
    float dxv = XV(idx + 1) - XV(idx);
    float two_c   =  6.0f * dxv - 4.0f * y0 - 2.0f * y1;
    float three_d = -6.0f * dxv + 3.0f * (y0 + y1);
    dXg[(size_t)k * (BB * CC) + sid] = y0 + (two_c + three_d * frac) * frac;
  }
#undef XV
}

// ---------------------------------------------------------------------------
// Kernel 2: persistent RK4 integrator.  32 WGs; WG g owns h-columns
// [4g, 4g+4) and the matching 256-column slice of W2 pinned in LDS in
// WMMA-fragment order.  W1 B-fragments live in registers for the whole
// kernel; GEMM2 A-fragments are register-cached across its 4 N-tiles.
// One device-scope barrier per vf evaluation.
// ---------------------------------------------------------------------------
__global__ __launch_bounds__(NTHR, 1) void cde_main(
    const float* __restrict__ z0, const float* __restrict__ W1,
    const float* __restrict__ b1, const float* __restrict__ W2,
    const float* __restrict__ b2, const float* __restrict__ Wd,
    const float* __restrict__ bd, const float* __restrict__ dXg,
    float* __restrict__ gzin,  float* __restrict__ gzfin,
    unsigned* __restrict__ cnt, unsigned* __restrict__ gen,
    float* __restrict__ out)
{
  // LDS: 128K (W2 slice) + 2x16.9K (zin,h) + k/z slices = ~164 KB
  __shared__ float W2F[16 * 32 * 32 * 2];   // [ntile][kblk4][lane][2]
  __shared__ float zbuf[BB * HSTR];
  __shared__ float hbuf[BB * HSTR];
  __shared__ float zloc[BB * HL];
  __shared__ float kbuf[4][BB * HL];
  __shared__ unsigned sgen;

  const int tid = threadIdx.x;
  const int wg  = blockIdx.x;
  const int n0  = wg * NCOLS;
  const int h0  = wg * HL;
  const int w   = tid >> 5;       // wave id 0..7
  const int l   = tid & 31;       // lane id
  const int lhi = l >> 4;         // half-wave select
  const int llo = l & 15;

  if (tid == 0) sgen = 0;

  // --- pin W2 slice in LDS, pre-swizzled to B-fragment order -------------
  for (int idx = tid; idx < 16 * 32 * 32; idx += NTHR) {
    int nt2 = idx >> 10;
    int kb4 = (idx >> 5) & 31;
    int ln  = idx & 31;
    int krow = kb4 * 4 + 2 * (ln >> 4);
    int col  = n0 + nt2 * 16 + (ln & 15);
    W2F[idx * 2 + 0] = W2[(size_t)(krow + 0) * (HH * CC) + col];
    W2F[idx * 2 + 1] = W2[(size_t)(krow + 1) * (HH * CC) + col];
  }
  // --- W1 B-fragments for this wave's N-tile: registers, whole kernel ----
  v2f w1b[32];
  {
    const int col = w * 16 + llo;
    #pragma unroll
    for (int kb4 = 0; kb4 < 32; ++kb4) {
      int krow = kb4 * 4 + 2 * lhi;
      v2f t;
      t.x = W1[(krow + 0) * HH + col];
      t.y = W1[(krow + 1) * HH + col];
      w1b[kb4] = t;
    }
  }
  if (tid < BB * HL) {
    int b = tid >> 2, hl = tid & 3;
    zloc[tid] = z0[b * HH + h0 + hl];
  }
  const float b1r = b1[w * 16 + llo];
  const int mt = w & 1;           // GEMM2 M-tile owned by this wave
  const int hw = w >> 1;          // GEMM2 local h-column owned by this wave
  float b2r[4];
  #pragma unroll
  for (int t4 = 0; t4 < 4; ++t4)
    b2r[t4] = b2[n0 + (hw * 4 + t4) * 16 + llo];
  __syncthreads();

  // --- device-scope barrier (arrive + generation, s_sleep spin) ----------
  auto gbar = [&]() {
    __threadfence();
    __syncthreads();
    if (threadIdx.x == 0) {
      unsigned target = ++sgen;
      unsigned prev = __hip_atomic_fetch_add(cnt, 1u, __ATOMIC_ACQ_REL,
                                             __HIP_MEMORY_SCOPE_AGENT);
      if (prev == (unsigned)(NBLK - 1)) {
        __hip_atomic_store(cnt, 0u, __ATOMIC_RELAXED, __HIP_MEMORY_SCOPE_AGENT);
        __hip_atomic_fetch_add(gen, 1u, __ATOMIC_RELEASE,
                               __HIP_MEMORY_SCOPE_AGENT);
      } else {
        while (__hip_atomic_load(gen, __ATOMIC_ACQUIRE,
                                 __HIP_MEMORY_SCOPE_AGENT) < target)
          __builtin_amdgcn_s_sleep(2);
      }
    }
    __syncthreads();
    __threadfence();
  };

  const float dt = 127.0f / 64.0f;
  for (int i = 0; i < NSTEPS; ++i) {
    for (int s = 0; s < 4; ++s) {
      const int g  = i * 4 + s;
      const int p  = g & 1;                              // zin double buffer
      const int kt = 2 * i + ((s == 0) ? 0 : ((s == 3) ? 2 : 1));
      // (1) publish this WG's zin slice
      if (tid < BB * HL) {
        float v = zloc[tid];
        if      (s == 1) v += 0.5f * dt * kbuf[0][tid];
        else if (s == 2) v += 0.5f * dt * kbuf[1][tid];
        else if (s == 3) v += dt * kbuf[2][tid];
        int b = tid >> 2, hl = tid & 3;
        gzin[p * (BB * HH) + b * HH + h0 + hl] = v;
      }
      gbar();
      // (2) stage full zin into padded LDS
      for (int idx = tid; idx < BB * HH; idx += NTHR)
        zbuf[(idx >> 7) * HSTR + (idx & 127)] = gzin[p * (BB * HH) + idx];
      __syncthreads();
      // (3) GEMM1: h = relu(zin @ W1 + b1)  (wave w owns N-tile w; B cached)
      for (int m2 = 0; m2 < 2; ++m2) {
        v8f acc = {};
        const float* ap = &zbuf[(m2 * 16 + llo) * HSTR + 2 * lhi];
        #pragma unroll
        for (int kb4 = 0; kb4 < 32; ++kb4) {
          v2f a = *(const v2f*)(ap + kb4 * 4);
          acc = wmma4(a, w1b[kb4], acc);
        }
        #pragma unroll
        for (int r = 0; r < 8; ++r) {
          float hv = acc[r] + b1r;
          hbuf[(m2 * 16 + r + 8 * lhi) * HSTR + w * 16 + llo] =
              fmaxf(hv, 0.0f);
        }
      }
      __syncthreads();
      // (4) GEMM2 slice + tanh + einsum contraction over C -> k slice.
      //     A-fragments identical across the 4 N-tiles -> register-cache.
      v2f afr[32];
      {
        const float* ap = &hbuf[(mt * 16 + llo) * HSTR + 2 * lhi];
        #pragma unroll
        for (int kb4 = 0; kb4 < 32; ++kb4)
          afr[kb4] = *(const v2f*)(ap + kb4 * 4);
      }
      float ks[8];
      #pragma unroll
      for (int r = 0; r < 8; ++r) ks[r] = 0.0f;
      const float* dxk = dXg + (size_t)kt * (BB * CC);
      for (int t4 = 0; t4 < 4; ++t4) {
        v8f acc = {};
        const float* bp = &W2F[(((hw * 4 + t4) * 32) * 32 + l) * 2];
        #pragma unroll
        for (int kb4 = 0; kb4 < 32; ++kb4) {
          v2f b = *(const v2f*)(bp + kb4 * 64);
          acc = wmma4(afr[kb4], b, acc);
        }
        #pragma unroll
        for (int r = 0; r < 8; ++r) {
          int m = mt * 16 + r + 8 * lhi;
          float f = tanhf(acc[r] + b2r[t4]);
          ks[r] += f * dxk[m * CC + t4 * 16 + llo];
        }
      }
      #pragma unroll
      for (int r = 0; r < 8; ++r) {     // reduce across the 16 N-lanes
        float v = ks[r];
        v += __shfl_xor(v, 8, 16);
        v += __shfl_xor(v, 4, 16);
        v += __shfl_xor(v, 2, 16);
        v += __shfl_xor(v, 1, 16);
        if (llo == 0) {
          int m = mt * 16 + r + 8 * lhi;
          kbuf[s][m * HL + hw] = v;
        }
      }
      __syncthreads();
    }
    // RK4 state update (WG-local slice)
    if (tid < BB * HL)
      zloc[tid] += (dt / 6.0f) * (kbuf[0][tid] +
                                  2.0f * (kbuf[1][tid] + kbuf[2][tid]) +
                                  kbuf[3][tid]);
    __syncthreads();
  }

  // --- readout: zT @ Wd + bd ---------------------------------------------
  if (tid < BB * HL) {
    int b = tid >> 2, hl = tid & 3;
    gzfin[b * HH + h0 + hl] = zloc[tid];
  }
  gbar();
  if (wg == 0) {
    for (int idx = tid; idx < BB * OUTN; idx += NTHR) {
      int b = idx >> 3, o = idx & 7;
      float acc = bd[o];
      for (int h = 0; h < HH; ++h)
        acc += gzfin[b * HH + h] * Wd[h * OUTN + o];
      out[idx] = acc;
    }
  }
}

// ---------------------------------------------------------------------------
extern "C" void kernel_launch(void* const* d_in, const int* in_sizes, int n_in,
                              void* d_out, int out_size, void* d_ws,
                              size_t ws_size, hipStream_t stream) {
  (void)in_sizes; (void)n_in; (void)out_size; (void)ws_size;
  const float* x  = (const float*)d_in[0];
  const float* z0 = (const float*)d_in[1];
  const float* W1 = (const float*)d_in[2];
  const float* b1 = (const float*)d_in[3];
  const float* W2 = (const float*)d_in[4];
  const float* b2 = (const float*)d_in[5];
  const float* Wd = (const float*)d_in[6];
  const float* bd = (const float*)d_in[7];
  float* out = (float*)d_out;

  float* ws    = (float*)d_ws;
  float* dXg   = ws;                       // [129][32][64]   = 1.03 MB
  float* gzin  = dXg + 129 * (BB * CC);    // [2][32][128]    = 32 KB
  float* gzfin = gzin + 2 * (BB * HH);     // [32][128]       = 16 KB
  unsigned* syncv = (unsigned*)(gzfin + BB * HH);   // {cnt, gen}

  cde_init  <<<1, 32, 0, stream>>>(syncv);
  cde_spline<<<8, 256, 0, stream>>>(x, dXg);
  cde_main  <<<NBLK, NTHR, 0, stream>>>(z0, W1, b1, W2, b2, Wd, bd, dXg,
                                        gzin, gzfin, syncv, syncv + 1, out);
}